// Rocket_27882927686329
// MI455X (gfx1250) — compile-verified
//
#include <hip/hip_runtime.h>

// ---------------------------------------------------------------------------
// ROCKET massive-conv for MI455X (gfx1250), compile-loop round 3.
// Geometry (fixed by reference): B=8, C=3, L=512, 9 taps, 84 kernels,
// 39 dilation blocks, O=3276, dense W width 567, pad 252, L_out 450.
// Per (b*c, f):  out(84x450) = Taps(84x10) * [Xgather(9x450); ones(1x450)]
// via V_WMMA_F32_16X16X4_F32, K padded to 12, bias folded as tap 9.
// A fragments register-resident; B fragments from zero-haloed LDS, shared
// across the 6 M-tiles; output written with non-temporal streaming stores.
// Main nt-loop is fully predicate-free; only nt==28 (wave 4) is masked.
// ---------------------------------------------------------------------------

typedef __attribute__((ext_vector_type(2))) float v2f;
typedef __attribute__((ext_vector_type(8))) float v8f;

#define NKER   84
#define KLEN_  9
#define LIN    512
#define LOUT   450
#define NFEAT  39
#define OCH    (NKER*NFEAT)   // 3276
#define KW     567            // 9 * 63
#define MAXPAD 252            // 4 * max_dilation
#define NBC    24             // B * C
#define MT     6              // ceil(84/16)  -> padded rows 96
#define NT     29             // ceil(450/16)
#define MROWS  96
#define XHALO  252
#define XTOT   1536           // covers index 252 + 463 + 7*63 = 1156

// --- Prep: recover dilation + 9 taps per channel from dense W rows. --------
__global__ void rocket_prep(const float* __restrict__ W,
                            float* __restrict__ taps,   // (3276, 9)
                            int*   __restrict__ dil) {  // (39)
    int o = blockIdx.x * blockDim.x + threadIdx.x;
    if (o >= OCH) return;
    const float* row = W + (size_t)o * KW;
    int j0 = 0;                                  // first nonzero = 252 - 4d
    while (j0 < KW - KLEN_ && row[j0] == 0.0f) ++j0;
    int d = (MAXPAD - j0) >> 2;
    if (d < 1) d = 1;
#pragma unroll
    for (int m = 0; m < KLEN_; ++m)
        taps[o * KLEN_ + m] = row[j0 + m * d];
    if ((o % NKER) == 0) dil[o / NKER] = d;
}

// --- One nt-column group: 6 M-tiles of 3 chained WMMAs + streaming stores. --
template <bool TAIL>
__device__ __forceinline__ void do_tiles(const v2f aa[MT][3], const v2f bb[3],
                                         float* __restrict__ obase,
                                         int t0, int g) {
    const bool ok = TAIL ? (t0 < LOUT) : true;
#pragma unroll
    for (int mt = 0; mt < MT; ++mt) {
        v8f c = {0.f, 0.f, 0.f, 0.f, 0.f, 0.f, 0.f, 0.f};
#pragma unroll
        for (int s = 0; s < 3; ++s)
            c = __builtin_amdgcn_wmma_f32_16x16x4_f32(
                    false, aa[mt][s], false, bb[s],
                    (short)0, c, false, false);

        float* p = obase + ((size_t)(mt * 16 + 8 * g)) * LOUT + t0;
        if (mt < MT - 1) {
            if (ok) {
#pragma unroll
                for (int r = 0; r < 8; ++r)
                    __builtin_nontemporal_store(c[r], p + (size_t)r * LOUT);
            }
        } else {
            if (ok && g == 0) {                 // rows 80..83 only
#pragma unroll
                for (int r = 0; r < 4; ++r)
                    __builtin_nontemporal_store(c[r], p + (size_t)r * LOUT);
            }
        }
    }
}

// --- Main: WMMA GEMM per (b*c, f) block; 8 waves of 32 (wave32). -----------
__global__ void __launch_bounds__(256)
rocket_wmma(const float* __restrict__ X,     // (24, 512)
            const float* __restrict__ bias,  // (3276)
            const float* __restrict__ taps,  // (3276, 9)
            const int*   __restrict__ dil,   // (39)
            float* __restrict__ out)         // (24, 3276, 450)
{
    __shared__ __align__(16) float sTap[MROWS][12];  // rows>=84, cols>=10 zero
    __shared__ __align__(16) float sXh[XTOT];        // zero halo both sides

    const int bc  = blockIdx.x % NBC;
    const int f   = blockIdx.x / NBC;
    const int tid = threadIdx.x;

    // Stage taps (+ bias as tap 9) with zero padding.
    for (int i = tid; i < MROWS * 12; i += 256) {
        int r = i / 12, cc = i % 12;
        float v = 0.0f;
        if (r < NKER) {
            if (cc < KLEN_)       v = taps[(f * NKER + r) * KLEN_ + cc];
            else if (cc == KLEN_) v = bias[f * NKER + r];
        }
        sTap[r][cc] = v;
    }
    // Stage X row with zero halo: sXh[252 + x] = X[x].
    for (int i = tid; i < XTOT; i += 256) {
        int xi = i - XHALO;
        sXh[i] = (xi >= 0 && xi < LIN) ? X[bc * LIN + xi] : 0.0f;
    }
    __syncthreads();

    const int d    = dil[f];
    const int wave = tid >> 5;
    const int lane = tid & 31;
    const int g    = lane >> 4;     // half-group: K = 4s + 2g + j
    const int hl   = lane & 15;

    // Preload all A fragments (nt-invariant), ds_load_b64 each.
    v2f aa[MT][3];
#pragma unroll
    for (int mt = 0; mt < MT; ++mt) {
#pragma unroll
        for (int s = 0; s < 3; ++s)
            aa[mt][s] = *(const v2f*)&sTap[mt * 16 + hl][4 * s + 2 * g];
    }

    float* const obase = out + ((size_t)bc * OCH + (size_t)f * NKER) * LOUT;

    // B-fragment gather offsets (uniform per half-group): (m-4)*d, m = 4s+2g+j.
    // K=9 (bias tap) forced to 1.0 in g==0's s==2,j==1 slot; K>=10 slots are
    // dead (matching A columns are zero) and may hold any finite halo value.
    auto load_b = [&](int t0, v2f bb[3]) {
#pragma unroll
        for (int s = 0; s < 3; ++s) {
#pragma unroll
            for (int j = 0; j < 2; ++j) {
                int m = 4 * s + 2 * g + j;
                bb[s][j] = sXh[XHALO + t0 + (m - 4) * d];
            }
        }
        bb[2].y = (g == 0) ? 1.0f : bb[2].y;
    };

    // Main path: nt = 0..27 -> every t0 = nt*16+hl <= 447 < 450, no masking.
    for (int nt = wave; nt < NT - 1; nt += 8) {
        const int t0 = nt * 16 + hl;
        v2f bb[3];
        load_b(t0, bb);
        do_tiles<false>(aa, bb, obase, t0, g);
    }
    // Tail: nt == 28 (owned by wave 4), columns 448..463 partially valid.
    if (wave == (NT - 1) % 8) {
        const int t0 = (NT - 1) * 16 + hl;
        v2f bb[3];
        load_b(t0, bb);
        do_tiles<true>(aa, bb, obase, t0, g);
    }
}

extern "C" void kernel_launch(void* const* d_in, const int* in_sizes, int n_in,
                              void* d_out, int out_size, void* d_ws, size_t ws_size,
                              hipStream_t stream) {
    const float* X = (const float*)d_in[0];   // (8,3,512)
    const float* W = (const float*)d_in[1];   // (3276,1,567)
    const float* b = (const float*)d_in[2];   // (3276,)
    float* out = (float*)d_out;

    float* taps = (float*)d_ws;                                   // 3276*9 f32
    int*   dil  = (int*)((char*)d_ws + (size_t)OCH * KLEN_ * sizeof(float));

    rocket_prep<<<(OCH + 255) / 256, 256, 0, stream>>>(W, taps, dil);
    rocket_wmma<<<NBC * NFEAT, 256, 0, stream>>>(X, b, taps, dil, out);
}